// MambaMelKWS_21973052686684
// MI455X (gfx1250) — compile-verified
//
#include <hip/hip_runtime.h>
#include <cstdint>
#include <cstddef>

// ---------------- model dims ----------------
#define B_      8
#define T_      512
#define F_      128
#define T2_     256          // T/2 after pooling
#define DM_     128          // d_model
#define DI_     256          // d_inner
#define DSTATE_ 16
#define DR_     8            // dt_rank
#define NL_     10
#define NC_     35
#define FLAT_   2048         // 64 * F/4
#define BT_     2048         // B * T2

typedef float v2f __attribute__((ext_vector_type(2)));
typedef float v8f __attribute__((ext_vector_type(8)));

__device__ __forceinline__ float silu_f(float x)     { return x / (1.f + __expf(-x)); }
__device__ __forceinline__ float softplus_f(float x) { return (x > 20.f) ? x : log1pf(__expf(x)); }

// ---- x [B,T,F] -> xT [B,1,F,T] (NCHW for convs) ---------------------------
__global__ void k_transpose_x(const float* __restrict__ x, float* __restrict__ xT)
{
    int idx = blockIdx.x * blockDim.x + threadIdx.x;
    if (idx >= B_ * F_ * T_) return;
    int t = idx & (T_ - 1);
    int f = (idx >> 9) & (F_ - 1);
    int b = idx >> 16;
    xT[idx] = x[((size_t)b * T_ + t) * F_ + f];
}

// ---- generic 3x3 SAME conv + folded BN + SiLU, NCHW -----------------------
__global__ void k_conv3x3(const float* __restrict__ in, const float* __restrict__ w,
                          const float* __restrict__ cbias, const float* __restrict__ bns,
                          const float* __restrict__ bnb, float* __restrict__ out,
                          int Cin, int Cout, int H, int W)
{
    size_t idx = (size_t)blockIdx.x * blockDim.x + threadIdx.x;
    size_t total = (size_t)B_ * Cout * H * W;
    if (idx >= total) return;
    int xw = (int)(idx % W); size_t r = idx / W;
    int yh = (int)(r % H);   r /= H;
    int co = (int)(r % Cout);
    int b  = (int)(r / Cout);
    float acc = cbias[co];
    const float* wbase = w  + (size_t)co * Cin * 9;
    const float* ibase = in + (size_t)b  * Cin * H * W;
    for (int ci = 0; ci < Cin; ++ci) {
        const float* ip = ibase + (size_t)ci * H * W;
        const float* wp = wbase + ci * 9;
        #pragma unroll
        for (int ky = 0; ky < 3; ++ky) {
            int yy = yh + ky - 1;
            if (yy < 0 || yy >= H) continue;
            #pragma unroll
            for (int kx = 0; kx < 3; ++kx) {
                int xx = xw + kx - 1;
                if (xx < 0 || xx >= W) continue;
                acc += ip[(size_t)yy * W + xx] * wp[ky * 3 + kx];
            }
        }
    }
    float v = acc * bns[co] + bnb[co];
    out[idx] = silu_f(v);
}

// ---- max pool (kh,kw) non-overlapping -------------------------------------
__global__ void k_maxpool(const float* __restrict__ in, float* __restrict__ out,
                          int C, int H, int W, int kh, int kw)
{
    int Ho = H / kh, Wo = W / kw;
    size_t idx = (size_t)blockIdx.x * blockDim.x + threadIdx.x;
    size_t total = (size_t)B_ * C * Ho * Wo;
    if (idx >= total) return;
    int ow = (int)(idx % Wo); size_t r = idx / Wo;
    int oh = (int)(r % Ho);   r /= Ho;
    int c  = (int)(r % C);
    int b  = (int)(r / C);
    const float* ip = in + ((size_t)b * C + c) * H * W;
    float m = -3.402823466e38f;
    for (int i = 0; i < kh; ++i)
        for (int j = 0; j < kw; ++j) {
            float v = ip[(size_t)(oh * kh + i) * W + (ow * kw + j)];
            m = v > m ? v : m;
        }
    out[idx] = m;
}

// ---- pool2 [B,64,32,T2] -> hseq [B,T2,FLAT]  (f = c*32 + hh) --------------
__global__ void k_reshape(const float* __restrict__ p2, float* __restrict__ hseq)
{
    int idx = blockIdx.x * blockDim.x + threadIdx.x;   // B*T2*FLAT = 4194304
    if (idx >= B_ * T2_ * FLAT_) return;
    int f = idx & (FLAT_ - 1);
    int t = (idx >> 11) & (T2_ - 1);
    int b = idx >> 19;
    int c = f >> 5, hh = f & 31;
    hseq[idx] = p2[(((size_t)b * 64 + c) * 32 + hh) * T2_ + t];
}

// ---- fp32 WMMA GEMM: C[M,N] (+)= A[M,K] * W[N,K]^T (+ bias) ---------------
// flags bit0: accumulate into C, bit1: add bias.
// One wave owns a 16x32 C tile (two 16x16 WMMA accumulators sharing the A
// fragment). OOB columns: row pointer clamped to N-1 for loads (branch-free;
// garbage lands in columns that are never stored), so the hot loop is
// 3x global_load_b64 + 2x v_wmma_f32_16x16x4_f32 with full EXEC throughout.
__global__ void k_gemm_wmma(const float* __restrict__ A, int lda,
                            const float* __restrict__ W,
                            const float* __restrict__ bias,
                            float* __restrict__ C, int ldc,
                            int M, int N, int K, int flags)
{
    const int lane = threadIdx.x;          // 0..31 (wave32)
    const int wave = threadIdx.y;          // 0..3
    const int half = lane >> 4;
    const int m16  = lane & 15;
    const int row0 = (blockIdx.y * 4 + wave) * 16;
    const int col0 = blockIdx.x * 32;
    if (row0 >= M) return;                 // wave-uniform: EXEC stays full
    const int n0  = col0 + m16;
    const int n1  = col0 + 16 + m16;
    const int n0c = (n0 < N) ? n0 : (N - 1);   // clamp for loads only
    const int n1c = (n1 < N) ? n1 : (N - 1);

    // A 16x4 fp32 fragment: lanes 0-15 -> K=k,k+1 ; lanes 16-31 -> K=k+2,k+3.
    // B mirrors A (K-major per lane). All base offsets even -> b64 aligned.
    const float* arow  = A + (size_t)(row0 + m16) * lda + 2 * half;
    const float* wrow0 = W + (size_t)n0c * K + 2 * half;
    const float* wrow1 = W + (size_t)n1c * K + 2 * half;

    v8f acc0 = {}, acc1 = {};
    for (int k = 0; k < K; k += 4) {
        if ((k & 127) == 0) {                       // scalar-uniform branch
            __builtin_prefetch(arow  + k + 128, 0, 0);   // global_prefetch_b8
            __builtin_prefetch(wrow0 + k + 128, 0, 0);
            __builtin_prefetch(wrow1 + k + 128, 0, 0);
        }
        v2f a  = *(const v2f*)(arow  + k);
        v2f b0 = *(const v2f*)(wrow0 + k);
        v2f b1 = *(const v2f*)(wrow1 + k);
        acc0 = __builtin_amdgcn_wmma_f32_16x16x4_f32(
                   false, a, false, b0, (short)0, acc0, false, false);
        acc1 = __builtin_amdgcn_wmma_f32_16x16x4_f32(
                   false, a, false, b1, (short)0, acc1, false, false);
    }

    float bv0 = 0.f, bv1 = 0.f;
    if (flags & 2) {
        if (n0 < N) bv0 = bias[n0];
        if (n1 < N) bv1 = bias[n1];
    }
    #pragma unroll
    for (int r = 0; r < 8; ++r) {
        int rr = row0 + half * 8 + r;      // lanes 0-15: M=r ; lanes 16-31: M=8+r
        size_t o = (size_t)rr * ldc;
        if (n0 < N) {
            float v = acc0[r] + bv0;
            if (flags & 1) v += C[o + n0];
            C[o + n0] = v;
        }
        if (n1 < N) {
            float v = acc1[r] + bv1;
            if (flags & 1) v += C[o + n1];
            C[o + n1] = v;
        }
    }
}

// ---- LayerNorm over 128 features, optional SiLU (one wave32 per row) ------
__global__ void k_ln(const float* __restrict__ in, float* __restrict__ out,
                     const float* __restrict__ s, const float* __restrict__ b,
                     int dosilu, int nrows)
{
    int row  = blockIdx.x * blockDim.y + threadIdx.y;
    int lane = threadIdx.x;                // 0..31
    if (row >= nrows) return;
    const float* ip = in + (size_t)row * DM_;
    float v[4];
    float sum = 0.f;
    #pragma unroll
    for (int j = 0; j < 4; ++j) { v[j] = ip[lane + 32 * j]; sum += v[j]; }
    #pragma unroll
    for (int off = 16; off; off >>= 1) sum += __shfl_xor(sum, off, 32);
    float mean = sum * (1.f / DM_);
    float vs = 0.f;
    #pragma unroll
    for (int j = 0; j < 4; ++j) { v[j] -= mean; vs += v[j] * v[j]; }
    #pragma unroll
    for (int off = 16; off; off >>= 1) vs += __shfl_xor(vs, off, 32);
    float inv = rsqrtf(vs * (1.f / DM_) + 1e-5f);
    float* op = out + (size_t)row * DM_;
    #pragma unroll
    for (int j = 0; j < 4; ++j) {
        float y = v[j] * inv * s[lane + 32 * j] + b[lane + 32 * j];
        if (dosilu) y = silu_f(y);
        op[lane + 32 * j] = y;
    }
}

// ---- causal depthwise conv1d (K=4, left pad 3) + SiLU ---------------------
__global__ void k_dwconv_silu(const float* __restrict__ xz, const float* __restrict__ cw,
                              const float* __restrict__ cb, float* __restrict__ xc)
{
    int idx = blockIdx.x * blockDim.x + threadIdx.x;   // BT*DI = 524288
    if (idx >= BT_ * DI_) return;
    int d  = idx & (DI_ - 1);
    int bt = idx >> 8;
    int t  = bt & (T2_ - 1);
    int b  = bt >> 8;
    float acc = cb[d];
    #pragma unroll
    for (int j = 0; j < 4; ++j) {
        int tt = t - 3 + j;
        if (tt >= 0) acc += xz[(size_t)(b * T2_ + tt) * (2 * DI_) + d] * cw[d * 4 + j];
    }
    xc[idx] = silu_f(acc);
}

// ---- sequential SSM scan: one thread per (b,d); 16 states in registers ----
__global__ void k_scan(const float* __restrict__ xz, const float* __restrict__ xc,
                       const float* __restrict__ dbc, const float* __restrict__ dtl,
                       const float* __restrict__ dt_b, const float* __restrict__ A_log,
                       const float* __restrict__ Dp, float* __restrict__ y)
{
    int b = blockIdx.x;       // 0..7
    int d = threadIdx.x;      // 0..255
    float Arow[DSTATE_], hst[DSTATE_];
    #pragma unroll
    for (int n = 0; n < DSTATE_; ++n) { Arow[n] = -__expf(A_log[d * DSTATE_ + n]); hst[n] = 0.f; }
    const float dtb = dt_b[d];
    const float Dpv = Dp[d];
    for (int t = 0; t < T2_; ++t) {
        size_t bt = (size_t)b * T2_ + t;
        float dtv = softplus_f(dtl[bt * DI_ + d] + dtb);
        float xcv = xc[bt * DI_ + d];
        float zv  = xz[bt * (2 * DI_) + DI_ + d];
        const float* bc = dbc + bt * (DR_ + 2 * DSTATE_);   // [dt(8) | B(16) | C(16)]
        float dBu = dtv * xcv;
        float acc = 0.f;
        #pragma unroll
        for (int n = 0; n < DSTATE_; ++n) {
            float dA = __expf(dtv * Arow[n]);
            hst[n] = dA * hst[n] + dBu * bc[DR_ + n];
            acc += hst[n] * bc[DR_ + DSTATE_ + n];
        }
        y[bt * DI_ + d] = (acc + Dpv * xcv) * silu_f(zv);
    }
}

// ---- masked mean pool over valid timesteps --------------------------------
__global__ void k_maskpool(const float* __restrict__ hln, const int* __restrict__ lengths,
                           float* __restrict__ pooled)
{
    int b = blockIdx.x;       // 0..7
    int d = threadIdx.x;      // 0..127
    int tl = lengths[b] / 2; if (tl < 1) tl = 1;
    float s = 0.f;
    for (int t = 0; t < tl; ++t) s += hln[((size_t)b * T2_ + t) * DM_ + d];
    pooled[b * DM_ + d] = s / (float)tl;
}

// ---- classifier: 128 -> 64 (SiLU) -> 35 -----------------------------------
__global__ void k_classifier(const float* __restrict__ pooled,
                             const float* __restrict__ w1, const float* __restrict__ b1,
                             const float* __restrict__ w2, const float* __restrict__ b2,
                             float* __restrict__ out)
{
    __shared__ float hbuf[64];
    int b = blockIdx.x;
    int t = threadIdx.x;      // 128 threads
    if (t < 64) {
        float acc = b1[t];
        for (int d = 0; d < DM_; ++d) acc += pooled[b * DM_ + d] * w1[t * DM_ + d];
        hbuf[t] = silu_f(acc);
    }
    __syncthreads();
    if (t < NC_) {
        float acc = b2[t];
        for (int hh = 0; hh < 64; ++hh) acc += hbuf[hh] * w2[t * 64 + hh];
        out[b * NC_ + t] = acc;
    }
}

// ===========================================================================
extern "C" void kernel_launch(void* const* d_in, const int* in_sizes, int n_in,
                              void* d_out, int out_size, void* d_ws, size_t ws_size,
                              hipStream_t stream)
{
    (void)in_sizes; (void)n_in; (void)out_size; (void)ws_size;

    const float* x        = (const float*)d_in[0];
    const int*   lengths  = (const int*)  d_in[1];
    const float* conv_w1  = (const float*)d_in[2];
    const float* conv_b1  = (const float*)d_in[3];
    const float* bn1_s    = (const float*)d_in[4];
    const float* bn1_b    = (const float*)d_in[5];
    const float* conv_w2  = (const float*)d_in[6];
    const float* conv_b2  = (const float*)d_in[7];
    const float* bn2_s    = (const float*)d_in[8];
    const float* bn2_b    = (const float*)d_in[9];
    const float* conv_w3  = (const float*)d_in[10];
    const float* conv_b3  = (const float*)d_in[11];
    const float* bn3_s    = (const float*)d_in[12];
    const float* bn3_b    = (const float*)d_in[13];
    const float* conv_w4  = (const float*)d_in[14];
    const float* conv_b4  = (const float*)d_in[15];
    const float* bn4_s    = (const float*)d_in[16];
    const float* bn4_b    = (const float*)d_in[17];
    const float* proj_w   = (const float*)d_in[18];
    const float* proj_b   = (const float*)d_in[19];
    const float* proj_ln_s= (const float*)d_in[20];
    const float* proj_ln_b= (const float*)d_in[21];
    const float* blk_ln_s = (const float*)d_in[22];
    const float* blk_ln_b = (const float*)d_in[23];
    const float* in_w     = (const float*)d_in[24];
    const float* cw       = (const float*)d_in[25];
    const float* cb       = (const float*)d_in[26];
    const float* xp_w     = (const float*)d_in[27];
    const float* dt_w     = (const float*)d_in[28];
    const float* dt_b     = (const float*)d_in[29];
    const float* A_log    = (const float*)d_in[30];
    const float* Dp       = (const float*)d_in[31];
    const float* out_w    = (const float*)d_in[32];
    const float* pre_ln_s = (const float*)d_in[33];
    const float* pre_ln_b = (const float*)d_in[34];
    const float* cls_w1   = (const float*)d_in[35];
    const float* cls_b1   = (const float*)d_in[36];
    const float* cls_w2   = (const float*)d_in[37];
    const float* cls_b2   = (const float*)d_in[38];
    float* out = (float*)d_out;

    // ---- scratch layout (floats); two 64MB ping-pong buffers, heavy reuse --
    float* ws   = (float*)d_ws;
    float* xT   = ws;                       //   524288  [B,1,F,T]
    float* bufA = ws + 524288;              // 16777216
    float* bufB = ws + 17301504;            // 16777216  (total 34,078,720 f = 136 MB)
    float* p1   = bufA;                     // pool1 out  [B,32,64,256]
    float* c3   = bufB;                     // conv3 out  [B,64,64,256]
    float* c4   = bufA;                     // conv4 out
    float* p2   = bufB;                     // pool2 out  [B,64,32,256]
    float* hseq = bufA;                     // [BT, FLAT]
    float* h    = bufB;                     // [BT, DM]  (overwrites dead p2)
    float* xln  = bufA;                     // [BT, DM]  (overwrites dead hseq)
    float* xz   = bufA + 262144;            // [BT, 2*DI]
    float* xc   = bufA + 1310720;           // [BT, DI]
    float* dbc  = bufA + 1835008;           // [BT, 40]
    float* dtl  = bufA + 1916928;           // [BT, DI]
    float* yb   = bufA + 2441216;           // [BT, DI]
    float* pooled = xT;                     // reuse (convs done)  [B, DM]

    const int TPB = 256;
    dim3 gblk(32, 4, 1);                    // 4 waves, each owns a 16x32 C tile
    dim3 lblk(32, 8, 1);                    // LN: 8 rows per block, wave per row

    // ---- conv front-end ----------------------------------------------------
    k_transpose_x<<<(B_*F_*T_)/TPB, TPB, 0, stream>>>(x, xT);
    k_conv3x3<<<(8*32*128*512)/TPB, TPB, 0, stream>>>(xT,  conv_w1, conv_b1, bn1_s, bn1_b, bufA, 1,  32, 128, 512);
    k_conv3x3<<<(8*32*128*512)/TPB, TPB, 0, stream>>>(bufA,conv_w2, conv_b2, bn2_s, bn2_b, bufB, 32, 32, 128, 512);
    k_maxpool<<<(8*32*64*256)/TPB,  TPB, 0, stream>>>(bufB, p1, 32, 128, 512, 2, 2);
    k_conv3x3<<<(8*64*64*256)/TPB,  TPB, 0, stream>>>(p1,  conv_w3, conv_b3, bn3_s, bn3_b, c3,  32, 64, 64, 256);
    k_conv3x3<<<(8*64*64*256)/TPB,  TPB, 0, stream>>>(c3,  conv_w4, conv_b4, bn4_s, bn4_b, c4,  64, 64, 64, 256);
    k_maxpool<<<(8*64*32*256)/TPB,  TPB, 0, stream>>>(c4, p2, 64, 64, 256, 2, 1);
    k_reshape<<<(B_*T2_*FLAT_)/TPB, TPB, 0, stream>>>(p2, hseq);

    // ---- projection + LN + SiLU -------------------------------------------
    k_gemm_wmma<<<dim3(DM_/32, BT_/64), gblk, 0, stream>>>(hseq, FLAT_, proj_w, proj_b, h, DM_, BT_, DM_, FLAT_, 2);
    k_ln<<<BT_/8, lblk, 0, stream>>>(h, h, proj_ln_s, proj_ln_b, 1, BT_);

    // ---- 10 Mamba blocks ---------------------------------------------------
    for (int i = 0; i < NL_; ++i) {
        k_ln<<<BT_/8, lblk, 0, stream>>>(h, xln, blk_ln_s + i*DM_, blk_ln_b + i*DM_, 0, BT_);
        k_gemm_wmma<<<dim3((2*DI_)/32, BT_/64), gblk, 0, stream>>>(
            xln, DM_, in_w + (size_t)i*2*DI_*DM_, nullptr, xz, 2*DI_, BT_, 2*DI_, DM_, 0);
        k_dwconv_silu<<<(BT_*DI_)/TPB, TPB, 0, stream>>>(xz, cw + i*DI_*4, cb + i*DI_, xc);
        k_gemm_wmma<<<dim3(2, BT_/64), gblk, 0, stream>>>(                       // N=40 (clamped tail)
            xc, DI_, xp_w + (size_t)i*(DR_+2*DSTATE_)*DI_, nullptr, dbc, DR_+2*DSTATE_, BT_, DR_+2*DSTATE_, DI_, 0);
        k_gemm_wmma<<<dim3(DI_/32, BT_/64), gblk, 0, stream>>>(                  // dt: K=8, lda=40
            dbc, DR_+2*DSTATE_, dt_w + (size_t)i*DI_*DR_, nullptr, dtl, DI_, BT_, DI_, DR_, 0);
        k_scan<<<B_, DI_, 0, stream>>>(xz, xc, dbc, dtl, dt_b + i*DI_,
                                       A_log + (size_t)i*DI_*DSTATE_, Dp + i*DI_, yb);
        k_gemm_wmma<<<dim3(DM_/32, BT_/64), gblk, 0, stream>>>(                  // residual add
            yb, DI_, out_w + (size_t)i*DM_*DI_, nullptr, h, DM_, BT_, DM_, DI_, 1);
    }

    // ---- head --------------------------------------------------------------
    k_ln<<<BT_/8, lblk, 0, stream>>>(h, xln, pre_ln_s, pre_ln_b, 0, BT_);
    k_maskpool<<<B_, DM_, 0, stream>>>(xln, lengths, pooled);
    k_classifier<<<B_, 128, 0, stream>>>(pooled, cls_w1, cls_b1, cls_w2, cls_b2, out);
}